// TemporalMSDA_23175643529396
// MI455X (gfx1250) — compile-verified
//
#include <hip/hip_runtime.h>
#include <hip/hip_bf16.h>
#include <math.h>

// ---------------------------------------------------------------------------
// TemporalMSDA for MI455X (gfx1250, wave32).
// Pipeline:
//   0) pack weights into K-pair-interleaved layout (B-fragment = one b64 load)
//   1) hs = hidden + pos
//   2) value = mask * (enc @ Wv + bv)            (fp32 WMMA GEMM, M=245760 N=256)
//   3) locbuf = hs @ Wo + bo ; aw = hs @ Wa + ba (fp32 WMMA GEMM, N=128)
//   4) loc = ref + locbuf / T[l] (in place) ; softmax16(aw) (in place in d_out)
//   5) msda = deformable temporal sampling       (wave per (b,q,h), lane=channel)
//   6) out  = msda @ Wout + bout                 (fp32 WMMA GEMM, N=256)
// GEMM: each wave computes a 16(M) x 64(N) strip = 4 WMMA tiles sharing one
// A fragment per K-step -> 5 b64 loads per 4 v_wmma_f32_16x16x4_f32.
// ---------------------------------------------------------------------------

typedef __attribute__((ext_vector_type(2))) float v2f;
typedef __attribute__((ext_vector_type(8))) float v8f;

#define CB 32
#define CQ 1024
#define CD 256
#define CH 8
#define CL 4
#define CP 4
#define CS 7680
#define CDh 32

__constant__ int   kTS[4]  = {4096, 2048, 1024, 512};
__constant__ int   kLSI[4] = {0, 4096, 6144, 7168};
__constant__ float kInvT[4] = {1.0f/4096.0f, 1.0f/2048.0f, 1.0f/1024.0f, 1.0f/512.0f};

// ---------------------------------------------------------------------------
// hs = a + b
// ---------------------------------------------------------------------------
__global__ void add_pe_kernel(const float* __restrict__ a, const float* __restrict__ b,
                              float* __restrict__ c, int n) {
  int i = blockIdx.x * blockDim.x + threadIdx.x;
  if (i < n) c[i] = a[i] + b[i];
}

// ---------------------------------------------------------------------------
// Pack W[256 x N] -> Wp[((k/2)*N + n)*2 + (k&1)] so that a B fragment
// (K pair {k, k+1} at column n) is a single aligned 8-byte load.
// ---------------------------------------------------------------------------
__global__ void pack_weight_kernel(const float* __restrict__ W,
                                   float* __restrict__ Wp, int N) {
  int i = blockIdx.x * blockDim.x + threadIdx.x;   // over 256*N
  if (i >= 256 * N) return;
  const int k = i / N;
  const int n = i - k * N;
  Wp[((size_t)(k >> 1) * N + n) * 2 + (k & 1)] = W[i];
}

// ---------------------------------------------------------------------------
// C[M x N] = A[M x 256] @ W[256 x N] + bias (W pre-packed), optional row mask.
// Wave strip: 16 rows x 64 cols (4 tiles).  Fragments per CDNA5 ISA 7.12.2:
//   A 16x4 : lane(0..15)=row, vgpr{0,1}=K{0,1}, lane-half (kh) selects K+=2
//   B 4x16 : lane(0..15)=col, vgpr{0,1}=K{0,1}, lane-half selects K+=2
//   C 16x16: lane(0..15)=col, vgpr r -> row M=r, lane-half selects M+=8
// ---------------------------------------------------------------------------
template <bool MASKED>
__global__ void gemm256_wmma_f32(const float* __restrict__ A,
                                 const float* __restrict__ Wp,
                                 const float* __restrict__ bias,
                                 float* __restrict__ C,
                                 int M, int N,
                                 const unsigned char* __restrict__ mask) {
  const int lane = threadIdx.x & 31;
  const int wave = threadIdx.x >> 5;
  const int nStrips = N >> 6;                               // 64-wide strips
  const int totalStrips = (M >> 4) * nStrips;
  const int s = blockIdx.x * (blockDim.x >> 5) + wave;      // wave-uniform
  if (s >= totalStrips) return;                              // whole-wave exit

  const int m0 = (s / nStrips) << 4;
  const int n0 = (s % nStrips) << 6;
  const int c  = lane & 15;
  const int kh = lane >> 4;

  const float* __restrict__ aptr = A + (size_t)(m0 + c) * 256 + 2 * kh;
  // element index of packed B frag: ((k>>1) + kh)*N*2 + (n0 + 16j + c)*2
  const float* __restrict__ bp = Wp + ((size_t)kh * N + n0 + c) * 2;

  v8f acc0 = {0.f,0.f,0.f,0.f,0.f,0.f,0.f,0.f};
  v8f acc1 = acc0, acc2 = acc0, acc3 = acc0;

#pragma unroll 4
  for (int k = 0; k < 256; k += 4) {
    const v2f a = *reinterpret_cast<const v2f*>(aptr + k);  // K = k+2kh, +1
    const float* bk = bp + (size_t)(k >> 1) * (N * 2);
    const v2f b0 = *reinterpret_cast<const v2f*>(bk);
    const v2f b1 = *reinterpret_cast<const v2f*>(bk + 32);
    const v2f b2 = *reinterpret_cast<const v2f*>(bk + 64);
    const v2f b3 = *reinterpret_cast<const v2f*>(bk + 96);
    acc0 = __builtin_amdgcn_wmma_f32_16x16x4_f32(false, a, false, b0, (short)0, acc0, false, false);
    acc1 = __builtin_amdgcn_wmma_f32_16x16x4_f32(false, a, false, b1, (short)0, acc1, false, false);
    acc2 = __builtin_amdgcn_wmma_f32_16x16x4_f32(false, a, false, b2, (short)0, acc2, false, false);
    acc3 = __builtin_amdgcn_wmma_f32_16x16x4_f32(false, a, false, b3, (short)0, acc3, false, false);
  }

  // 8 mask bytes for rows m0+8*kh .. +7 in one aligned 8B load.
  unsigned long long mm = 0;
  if (MASKED)
    mm = *reinterpret_cast<const unsigned long long*>(mask + m0 + 8 * kh);

  const v8f accs[4] = {acc0, acc1, acc2, acc3};
#pragma unroll
  for (int j = 0; j < 4; ++j) {
    const int nj = n0 + 16 * j + c;
    const float bb = bias[nj];
#pragma unroll
    for (int r = 0; r < 8; ++r) {
      const int row = m0 + 8 * kh + r;
      float v = accs[j][r] + bb;
      if (MASKED) v = ((mm >> (8 * r)) & 0xffull) ? v : 0.0f;
      C[(size_t)row * N + nj] = v;
    }
  }
}

// ---------------------------------------------------------------------------
// locbuf[idx] = ref[b,q,l] + locbuf[idx] / T[l]   (idx over B*Q*H*L*P)
// ---------------------------------------------------------------------------
__global__ void make_loc_kernel(float* __restrict__ locbuf,
                                const float* __restrict__ ref, int n) {
  int i = blockIdx.x * blockDim.x + threadIdx.x;
  if (i >= n) return;
  const int l  = (i >> 2) & 3;       // P=4, L=4
  const int bq = i >> 7;             // / (H*L*P=128)
  locbuf[i] = ref[bq * CL + l] + locbuf[i] * kInvT[l];
}

// ---------------------------------------------------------------------------
// In-place softmax over 16 logits per (b,q,h).
// ---------------------------------------------------------------------------
__global__ void softmax16_kernel(float* __restrict__ aw, int nGroups) {
  int g = blockIdx.x * blockDim.x + threadIdx.x;
  if (g >= nGroups) return;
  float* p = aw + (size_t)g * 16;
  float m = p[0];
#pragma unroll
  for (int i = 1; i < 16; ++i) m = fmaxf(m, p[i]);
  float e[16];
  float ssum = 0.f;
#pragma unroll
  for (int i = 0; i < 16; ++i) { e[i] = expf(p[i] - m); ssum += e[i]; }
  const float inv = 1.0f / ssum;
#pragma unroll
  for (int i = 0; i < 16; ++i) p[i] = e[i] * inv;
}

// ---------------------------------------------------------------------------
// Deformable temporal sampling: one wave per (b,q,h), lane = channel (Dh=32).
// ---------------------------------------------------------------------------
__global__ void msda_sample_kernel(const float* __restrict__ value,
                                   const float* __restrict__ loc,
                                   const float* __restrict__ aw,
                                   float* __restrict__ msda) {
  const int lane = threadIdx.x & 31;
  const int wave = threadIdx.x >> 5;
  const int g = blockIdx.x * (blockDim.x >> 5) + wave;   // (b*Q+q)*H + h
  if (g >= CB * CQ * CH) return;
  const int h  = g & (CH - 1);
  const int bq = g >> 3;
  const int b  = bq >> 10;                               // Q=1024

  const float* __restrict__ vb = value + (size_t)b * CS * CD + h * CDh + lane;
  const int base16 = g * 16;
  float acc = 0.0f;

#pragma unroll
  for (int l = 0; l < CL; ++l) {
    const int Tl = kTS[l];
    const float* __restrict__ vlb = vb + (size_t)kLSI[l] * CD;
#pragma unroll
    for (int p = 0; p < CP; ++p) {
      const int idx = base16 + l * CP + p;
      const float x  = loc[idx] * (float)Tl - 0.5f;
      const float wA = aw[idx];
      const float x0 = floorf(x);
      const float w1 = x - x0;
      const int i0 = (int)x0;
      const int i1 = i0 + 1;
      const float v0 = (i0 >= 0 && i0 < Tl) ? vlb[(size_t)i0 * CD] : 0.0f;
      const float v1 = (i1 >= 0 && i1 < Tl) ? vlb[(size_t)i1 * CD] : 0.0f;
      acc += wA * (v0 * (1.0f - w1) + v1 * w1);
    }
  }
  msda[(size_t)bq * CD + h * CDh + lane] = acc;
}

// ---------------------------------------------------------------------------
extern "C" void kernel_launch(void* const* d_in, const int* in_sizes, int n_in,
                              void* d_out, int out_size, void* d_ws, size_t ws_size,
                              hipStream_t stream) {
  (void)in_sizes; (void)n_in; (void)out_size; (void)ws_size;

  const float* hidden = (const float*)d_in[0];          // [B,Q,D]
  const float* enc    = (const float*)d_in[1];          // [B,S,D]
  const unsigned char* mask = (const unsigned char*)d_in[2];  // [B,S] bool
  const float* pos    = (const float*)d_in[3];          // [B,Q,D]
  const float* ref    = (const float*)d_in[4];          // [B,Q,L,1]
  const float* vw     = (const float*)d_in[7];          // [D,D]
  const float* vbias  = (const float*)d_in[8];          // [D]
  const float* ow     = (const float*)d_in[9];          // [D,128]
  const float* obias  = (const float*)d_in[10];         // [128]
  const float* aww    = (const float*)d_in[11];         // [D,128]
  const float* awb    = (const float*)d_in[12];         // [128]
  const float* outw   = (const float*)d_in[13];         // [D,D]
  const float* outb   = (const float*)d_in[14];         // [D]

  constexpr int BQ    = CB * CQ;            // 32768
  constexpr int BQD   = BQ * CD;            // 8,388,608
  constexpr int BS    = CB * CS;            // 245,760
  constexpr size_t BSD = (size_t)BS * CD;   // 62,914,560
  constexpr int HLP   = CH * CL * CP;       // 128
  constexpr int BQHLP = BQ * HLP;           // 4,194,304
  constexpr int BQH   = BQ * CH;            // 262,144

  float* ws     = (float*)d_ws;
  float* hs     = ws;                        // BQD
  float* value  = hs + BQD;                  // BSD
  float* locbuf = value + BSD;               // BQHLP
  float* msda   = locbuf + BQHLP;            // BQD
  float* vwP    = msda + BQD;                // 65536
  float* owP    = vwP + 256 * 256;           // 32768
  float* awP    = owP + 256 * 128;           // 32768
  float* outwP  = awP + 256 * 128;           // 65536

  float* out0  = (float*)d_out;              // [B,Q,D]
  float* awout = out0 + BQD;                 // [B,Q,H,L,P]

  // 0) pack weights (K-pair interleave)
  pack_weight_kernel<<<(256 * 256 + 255) / 256, 256, 0, stream>>>(vw,   vwP,   256);
  pack_weight_kernel<<<(256 * 128 + 255) / 256, 256, 0, stream>>>(ow,   owP,   128);
  pack_weight_kernel<<<(256 * 128 + 255) / 256, 256, 0, stream>>>(aww,  awP,   128);
  pack_weight_kernel<<<(256 * 256 + 255) / 256, 256, 0, stream>>>(outw, outwP, 256);

  // 1) hs = hidden + pos
  add_pe_kernel<<<(BQD + 255) / 256, 256, 0, stream>>>(hidden, pos, hs, BQD);

  // 2) value = mask * (enc @ Wv + bv) : 15360*4 = 61440 strips, 8 waves/blk
  {
    const int strips = (BS / 16) * (CD / 64);
    gemm256_wmma_f32<true><<<(strips + 7) / 8, 256, 0, stream>>>(
        enc, vwP, vbias, value, BS, CD, mask);
  }

  // 3a) locbuf = hs @ offs_w + offs_b   (M=32768, N=128)
  {
    const int strips = (BQ / 16) * (HLP / 64);
    gemm256_wmma_f32<false><<<(strips + 7) / 8, 256, 0, stream>>>(
        hs, owP, obias, locbuf, BQ, HLP, nullptr);
  }
  // 3b) awout = hs @ attw_w + attw_b    (logits, softmaxed in place below)
  {
    const int strips = (BQ / 16) * (HLP / 64);
    gemm256_wmma_f32<false><<<(strips + 7) / 8, 256, 0, stream>>>(
        hs, awP, awb, awout, BQ, HLP, nullptr);
  }

  // 4) loc = ref + offs/T  ;  softmax16(aw)
  make_loc_kernel<<<(BQHLP + 255) / 256, 256, 0, stream>>>(locbuf, ref, BQHLP);
  softmax16_kernel<<<(BQH + 255) / 256, 256, 0, stream>>>(awout, BQH);

  // 5) deformable sampling: wave per (b,q,h), 8 waves/block
  msda_sample_kernel<<<(BQH + 7) / 8, 256, 0, stream>>>(value, locbuf, awout, msda);

  // 6) out = msda @ out_w + out_b
  {
    const int strips = (BQ / 16) * (CD / 64);
    gemm256_wmma_f32<false><<<(strips + 7) / 8, 256, 0, stream>>>(
        msda, outwP, outb, out0, BQ, CD, nullptr);
  }
}